// SplineBlock_52407190946008
// MI455X (gfx1250) — compile-verified
//
#include <hip/hip_runtime.h>
#include <math.h>

typedef __attribute__((ext_vector_type(2))) float v2f;
typedef __attribute__((ext_vector_type(8))) float v8f;

#define TPB 256

// ---------------------------------------------------------------------------
// Zero-fill
// ---------------------------------------------------------------------------
__global__ void zero_kernel(float* __restrict__ p, long long n) {
    long long i = (long long)blockIdx.x * TPB + threadIdx.x;
    long long stride = (long long)gridDim.x * TPB;
    for (; i < n; i += stride) p[i] = 0.0f;
}

// ---------------------------------------------------------------------------
// Degree (mean-aggregation denominator); depends only on dst, computed once.
// ---------------------------------------------------------------------------
__global__ void degree_kernel(const int* __restrict__ ei, float* __restrict__ deg, int E) {
    int e = blockIdx.x * TPB + threadIdx.x;
    if (e < E) atomicAdd(&deg[ei[E + e]], 1.0f);
}

// ---------------------------------------------------------------------------
// Pack B (= [w (27,K,Fout) ; root (K,Fout)]) into WMMA fragment order.
// For output tile t (16 consecutive columns of the 28*Fout-wide output),
// k-step k4, lane l (half = l>>4, lh = l&15):
//   Bpk[(t*NK4 + k4)*32 + l] = { B[4*k4+2*half    ][t*16+lh],
//                                B[4*k4+2*half + 1][t*16+lh] }     (0 past K)
// so the GEMM reads one coalesced float2 per lane per k-step, guard-free.
// ---------------------------------------------------------------------------
__global__ void pack_b_kernel(const float* __restrict__ W, const float* __restrict__ root,
                              v2f* __restrict__ Bpk, int K, int nk4, int Fout)
{
    const int ntiles = 28 * Fout / 16;
    const int total  = ntiles * nk4 * 32;
    int i = blockIdx.x * TPB + threadIdx.x;
    if (i >= total) return;

    const int t    = i / (nk4 * 32);
    const int rem  = i - t * nk4 * 32;
    const int k4   = rem >> 5;
    const int lane = rem & 31;
    const int half = lane >> 4;
    const int lh   = lane & 15;

    const int c0     = t * 16;
    const int kslice = c0 / Fout;              // Fout % 16 == 0 -> tile within one slice
    const int o      = c0 - kslice * Fout + lh;
    const float* B   = (kslice < 27) ? (W + kslice * K * Fout) : root;
    const int kb     = 4 * k4 + 2 * half;

    v2f v;
    v.x = (kb     < K) ? B[kb * Fout + o]       : 0.0f;
    v.y = (kb + 1 < K) ? B[(kb + 1) * Fout + o] : 0.0f;
    Bpk[i] = v;
}

// ---------------------------------------------------------------------------
// WMMA weight transform:  XW[n, k*Fout+o] = sum_i X[n,i] * W[k,i,o]
// (slot k==27 = root). f32 16x16x4 WMMA, KPAD compile-time for full unroll.
//
// Fragment layouts (CDNA5 ISA 7.12.2, wave32):
//   A (16x4 f32, v2f): lanes 0-15 -> M=lane,    K={4k,4k+1};
//                      lanes16-31 -> M=lane-16, K={4k+2,4k+3}
//   B (4x16 f32, v2f): b.x: K=4k+2*half ; b.y: K=4k+2*half+1 ; col = c0+lh
//   C/D (16x16, v8f):  VGPR r: lanes0-15 -> M=r, lanes16-31 -> M=r+8; col=c0+lh
// ---------------------------------------------------------------------------
template <int KPAD>
__global__ __launch_bounds__(TPB)
void spline_gemm_wmma(const float* __restrict__ X,    // (Nn, K) row-major
                      const v2f* __restrict__ Bpk,    // packed fragments
                      float* __restrict__ XW,         // (Nn, 28*Fout)
                      int K, int Fout)
{
    constexpr int NK4 = KPAD / 4;
    __shared__ float As[16 * 76];                 // 76 = padded row stride (bank-safe)
    const int tid = threadIdx.x;
    const int m0  = blockIdx.y * 16;
    const int ldx = 28 * Fout;

    // Stage the 16 x KPAD A tile into LDS, zero-padded past K (handles K=67).
    for (int i = tid; i < 16 * KPAD; i += TPB) {
        int m = i / KPAD;
        int k = i - m * KPAD;
        As[m * 76 + k] = (k < K) ? X[(m0 + m) * K + k] : 0.0f;
    }
    __syncthreads();

    const int wave = tid >> 5;
    const int lane = tid & 31;
    const int half = lane >> 4;
    const int lh   = lane & 15;

    const int tile = blockIdx.x * 8 + wave;       // global 16-column tile
    const v2f* Bt  = Bpk + tile * NK4 * 32 + lane;
    const float* Arow = &As[lh * 76 + half * 2];

    v8f acc = {};
#pragma unroll
    for (int k4 = 0; k4 < NK4; ++k4) {
        v2f a = *(const v2f*)(Arow + k4 * 4);     // 8B-aligned LDS read
        v2f b = Bt[k4 * 32];                      // coalesced 8B global read
        acc = __builtin_amdgcn_wmma_f32_16x16x4_f32(false, a, false, b,
                                                    (short)0, acc, false, false);
    }

    const int obase = (m0 + half * 8) * ldx + tile * 16 + lh;
#pragma unroll
    for (int r = 0; r < 8; ++r)
        XW[obase + r * ldx] = acc[r];
}

// ---------------------------------------------------------------------------
// Edge phase: one wave per edge. Recompute degree-1 closed B-spline basis
// (8 weights + 8 kernel indices) from edge_attr, gather-blend rows of XW at
// the source node, scatter-add into agg[dst]. Lanes cover Fout channels so
// each basis gather is a coalesced 128B read.
// ---------------------------------------------------------------------------
__global__ __launch_bounds__(TPB)
void edge_kernel(const float* __restrict__ XW, const int* __restrict__ ei,
                 const float* __restrict__ ea, float* __restrict__ agg,
                 int E, int Fout)
{
    const int wid  = blockIdx.x * (TPB / 32) + (threadIdx.x >> 5);
    const int lane = threadIdx.x & 31;
    if (wid >= E) return;

    const int src = ei[wid];
    const int dst = ei[E + wid];
    const int ldx = 28 * Fout;

    float v0 = ea[wid * 3 + 0] * 3.0f;
    float v1 = ea[wid * 3 + 1] * 3.0f;
    float v2 = ea[wid * 3 + 2] * 3.0f;
    int l0 = (int)v0; if (l0 > 2) l0 = 2;
    int l1 = (int)v1; if (l1 > 2) l1 = 2;
    int l2 = (int)v2; if (l2 > 2) l2 = 2;
    const float f0 = v0 - (float)l0, f1 = v1 - (float)l1, f2 = v2 - (float)l2;

    float wgt[8];
    int   idx[8];
#pragma unroll
    for (int s = 0; s < 8; ++s) {
        const int b0 = s & 1, b1 = (s >> 1) & 1, b2 = (s >> 2) & 1;
        wgt[s] = (b0 ? f0 : 1.0f - f0) * (b1 ? f1 : 1.0f - f1) * (b2 ? f2 : 1.0f - f2);
        idx[s] = ((l0 + b0) % 3) + ((l1 + b1) % 3) * 3 + ((l2 + b2) % 3) * 9;
    }

    const float* base = XW + src * ldx;
    for (int o = lane; o < Fout; o += 32) {
        float acc = 0.0f;
#pragma unroll
        for (int s = 0; s < 8; ++s)
            acc += wgt[s] * base[idx[s] * Fout + o];
        atomicAdd(&agg[dst * Fout + o], acc);
    }
}

// ---------------------------------------------------------------------------
// out[n, o] = agg[n,o]/max(deg[n],1) + XW[n, 27, o] (root term) + bias[o]
// ---------------------------------------------------------------------------
__global__ void finalize_kernel(const float* __restrict__ agg, const float* __restrict__ deg,
                                const float* __restrict__ XW, const float* __restrict__ bias,
                                float* __restrict__ out, int Nn, int Fout, int ldo)
{
    int i = blockIdx.x * TPB + threadIdx.x;
    if (i >= Nn * Fout) return;
    int n = i / Fout, o = i - n * Fout;
    float d = deg[n]; if (d < 1.0f) d = 1.0f;
    const int ldx = 28 * Fout;
    float v = agg[n * Fout + o] / d + XW[n * ldx + 27 * Fout + o] + bias[o];
    out[n * ldo + o] = v;
}

// ---------------------------------------------------------------------------
// BatchNorm statistics (biased var, matching jnp.var): one block per channel,
// emits fused scale/shift: y = x*scale[c] + shift[c].
// ---------------------------------------------------------------------------
__global__ __launch_bounds__(TPB)
void bn_stats_kernel(const float* __restrict__ h, int Nn, int C, int ld,
                     const float* __restrict__ gamma, const float* __restrict__ beta,
                     float* __restrict__ ss)   // ss[0..C)=scale, ss[C..2C)=shift
{
    __shared__ float r0[TPB], r1[TPB];
    const int c = blockIdx.x;
    float s = 0.0f, s2 = 0.0f;
    for (int n = threadIdx.x; n < Nn; n += TPB) {
        float v = h[n * ld + c];
        s += v; s2 += v * v;
    }
    r0[threadIdx.x] = s; r1[threadIdx.x] = s2;
    __syncthreads();
    for (int off = TPB / 2; off > 0; off >>= 1) {
        if (threadIdx.x < off) {
            r0[threadIdx.x] += r0[threadIdx.x + off];
            r1[threadIdx.x] += r1[threadIdx.x + off];
        }
        __syncthreads();
    }
    if (threadIdx.x == 0) {
        float m   = r0[0] / (float)Nn;
        float var = r1[0] / (float)Nn - m * m;
        float rs  = rsqrtf(var + 1e-5f);
        float sc  = gamma[c] * rs;
        ss[c]     = sc;
        ss[C + c] = beta[c] - m * sc;
    }
}

__global__ void bn_elu_kernel(float* __restrict__ h, int Nn, int C, int ld,
                              const float* __restrict__ ss)
{
    int i = blockIdx.x * TPB + threadIdx.x;
    if (i >= Nn * C) return;
    int n = i / C, c = i - n * C;
    float v = h[n * ld + c] * ss[c] + ss[C + c];
    h[n * ld + c] = (v > 0.0f) ? v : (expf(v) - 1.0f);
}

__global__ void pos_concat_kernel(const float* __restrict__ pos, float* __restrict__ h3,
                                  int Nn)
{
    int i = blockIdx.x * TPB + threadIdx.x;
    if (i >= Nn * 3) return;
    int n = i / 3, d = i - n * 3;
    h3[n * 67 + 64 + d] = pos[i];
}

// ---------------------------------------------------------------------------
// Orchestration
// ---------------------------------------------------------------------------
extern "C" void kernel_launch(void* const* d_in, const int* in_sizes, int n_in,
                              void* d_out, int out_size, void* d_ws, size_t ws_size,
                              hipStream_t stream)
{
    const float* x   = (const float*)d_in[0];   // (N,16)
    const float* pos = (const float*)d_in[1];   // (N,3)
    const int*   ei  = (const int*)  d_in[2];   // (2,E)
    const float* ea  = (const float*)d_in[3];   // (E,3)
    const float* w1  = (const float*)d_in[4];
    const float* r1  = (const float*)d_in[5];
    const float* b1  = (const float*)d_in[6];
    const float* g1  = (const float*)d_in[7];
    const float* be1 = (const float*)d_in[8];
    const float* w2  = (const float*)d_in[9];
    const float* r2  = (const float*)d_in[10];
    const float* b2  = (const float*)d_in[11];
    const float* g2  = (const float*)d_in[12];
    const float* be2 = (const float*)d_in[13];
    const float* w3  = (const float*)d_in[14];
    const float* r3  = (const float*)d_in[15];
    const float* b3  = (const float*)d_in[16];
    float* out = (float*)d_out;

    const int N = in_sizes[0] / 16;             // 50000 (multiple of 16)
    const int E = in_sizes[2] / 2;              // 800000

    // Workspace carve-up (floats)
    float* ws   = (float*)d_ws;
    float* xw   = ws;                                  // N * 28*64 (max across convs)
    float* agg  = xw  + (size_t)N * 28 * 64;           // N * 64
    float* deg  = agg + (size_t)N * 64;                // N
    float* h1   = deg + (size_t)N;                     // N * 32
    float* h3   = h1  + (size_t)N * 32;                // N * 67 (conv2 out ++ pos)
    float* ss   = h3  + (size_t)N * 67;                // 2 * 64
    v2f*   Bpk  = (v2f*)(ss + 128);                    // <= 56*17*32 v2f (~244 KB)

    const int eblocks = (E + (TPB / 32) - 1) / (TPB / 32);
    const int mtiles  = N / 16;

    // Degree (once; depends only on dst)
    zero_kernel<<<2048, TPB, 0, stream>>>(deg, (long long)N);
    degree_kernel<<<(E + TPB - 1) / TPB, TPB, 0, stream>>>(ei, deg, E);

    // ---- conv1: 16 -> 32 --------------------------------------------------
    {
        const int nk4 = 4, ntiles = 28 * 32 / 16, total = ntiles * nk4 * 32;
        pack_b_kernel<<<(total + TPB - 1) / TPB, TPB, 0, stream>>>(w1, r1, Bpk, 16, nk4, 32);
        spline_gemm_wmma<16><<<dim3(ntiles / 8, mtiles), TPB, 0, stream>>>(x, Bpk, xw, 16, 32);
    }
    zero_kernel<<<2048, TPB, 0, stream>>>(agg, (long long)N * 32);
    edge_kernel<<<eblocks, TPB, 0, stream>>>(xw, ei, ea, agg, E, 32);
    finalize_kernel<<<(N * 32 + TPB - 1) / TPB, TPB, 0, stream>>>(
        agg, deg, xw, b1, h1, N, 32, /*ldo=*/32);
    bn_stats_kernel<<<32, TPB, 0, stream>>>(h1, N, 32, 32, g1, be1, ss);
    bn_elu_kernel<<<(N * 32 + TPB - 1) / TPB, TPB, 0, stream>>>(h1, N, 32, 32, ss);

    // ---- conv2: 32 -> 64 (output into cols 0..63 of the 67-wide concat) ---
    {
        const int nk4 = 8, ntiles = 28 * 64 / 16, total = ntiles * nk4 * 32;
        pack_b_kernel<<<(total + TPB - 1) / TPB, TPB, 0, stream>>>(w2, r2, Bpk, 32, nk4, 64);
        spline_gemm_wmma<32><<<dim3(ntiles / 8, mtiles), TPB, 0, stream>>>(h1, Bpk, xw, 32, 64);
    }
    zero_kernel<<<2048, TPB, 0, stream>>>(agg, (long long)N * 64);
    edge_kernel<<<eblocks, TPB, 0, stream>>>(xw, ei, ea, agg, E, 64);
    finalize_kernel<<<(N * 64 + TPB - 1) / TPB, TPB, 0, stream>>>(
        agg, deg, xw, b2, h3, N, 64, /*ldo=*/67);
    bn_stats_kernel<<<64, TPB, 0, stream>>>(h3, N, 64, 67, g2, be2, ss);
    bn_elu_kernel<<<(N * 64 + TPB - 1) / TPB, TPB, 0, stream>>>(h3, N, 64, 67, ss);
    pos_concat_kernel<<<(N * 3 + TPB - 1) / TPB, TPB, 0, stream>>>(pos, h3, N);

    // ---- conv3: 67 -> 32 (no BN/ELU) --------------------------------------
    {
        const int nk4 = 17, ntiles = 28 * 32 / 16, total = ntiles * nk4 * 32;
        pack_b_kernel<<<(total + TPB - 1) / TPB, TPB, 0, stream>>>(w3, r3, Bpk, 67, nk4, 32);
        spline_gemm_wmma<68><<<dim3(ntiles / 8, mtiles), TPB, 0, stream>>>(h3, Bpk, xw, 67, 32);
    }
    zero_kernel<<<2048, TPB, 0, stream>>>(agg, (long long)N * 32);
    edge_kernel<<<eblocks, TPB, 0, stream>>>(xw, ei, ea, agg, E, 32);
    finalize_kernel<<<(N * 32 + TPB - 1) / TPB, TPB, 0, stream>>>(
        agg, deg, xw, b3, out, N, 32, /*ldo=*/32);
}